// MaxUnPool2DWithIndices_81260781240726
// MI455X (gfx1250) — compile-verified
//
#include <hip/hip_runtime.h>

// Max-unpool 2x2 / stride 2, NHWC, f32.
//   in  : (N, H, W, D)  f32
//   idx : (N, H, W, D)  i32 in {0,1,2,3};  p = di*2 + dj -> out (2h+di, 2w+dj)
//   out : (N, 2H, 2W, D) f32
//
// Pure streaming op: ~617 MB total traffic -> ~26.5 us roofline at 23.3 TB/s.
// One thread owns one float4 of channels for one pooled pixel and writes all
// four 16B slots of its 2x2 output block (one-hot select): full output
// coverage, no zero-init pass, no atomics. All loads/stores are 128-bit,
// lane-coalesced, and non-temporal (stream-once data; keep the 192MB L2 clean).
//
// Indexing comes entirely from the 3D launch grid -- no integer division:
//   threadIdx.x = d4 (0..63, float4 chunk of D=256)
//   threadIdx.y = w sub-index (0..3), blockIdx.x = w group (0..27)
//   blockIdx.y = h, blockIdx.z = n
// All offsets fit in 32 bits (max 102,760,448 elements).

typedef __attribute__((ext_vector_type(4))) float v4f;
typedef __attribute__((ext_vector_type(4))) int   v4i;

namespace {
constexpr int N  = 8;
constexpr int H  = 112;
constexpr int W  = 112;
constexpr int D  = 256;
constexpr int D4 = D / 4;          // 64 float4 chunks per pixel
constexpr int OH = 2 * H;
constexpr int OW = 2 * W;
}

__global__ __launch_bounds__(256)
void MaxUnPool2DWithIndices_81260781240726_kernel(const float* __restrict__ in,
                                                  const int*   __restrict__ idx,
                                                  float*       __restrict__ out)
{
    const int d4 = (int)threadIdx.x;                       // 0..63
    const int w  = (int)(blockIdx.x * 4 + threadIdx.y);    // 0..111
    const int h  = (int)blockIdx.y;                        // 0..111
    const int n  = (int)blockIdx.z;                        // 0..7

    const int ibase = (((n * H + h) * W + w) * D) + (d4 << 2);

    const v4f v  = __builtin_nontemporal_load((const v4f*)(in  + ibase));
    const v4i ix = __builtin_nontemporal_load((const v4i*)(idx + ibase));

    // Output block origin (2h, 2w); dj steps D floats, di steps OW*D floats.
    const int obase     = (((n * OH + 2 * h) * OW + 2 * w) * D) + (d4 << 2);
    const int rowStride = OW * D;                          // 57344 floats

#pragma unroll
    for (int p = 0; p < 4; ++p) {
        const int di = p >> 1;
        const int dj = p & 1;
        v4f o;
        o.x = (ix.x == p) ? v.x : 0.0f;
        o.y = (ix.y == p) ? v.y : 0.0f;
        o.z = (ix.z == p) ? v.z : 0.0f;
        o.w = (ix.w == p) ? v.w : 0.0f;
        __builtin_nontemporal_store(
            o, (v4f*)(out + obase + di * rowStride + dj * D));
    }
}

extern "C" void kernel_launch(void* const* d_in, const int* in_sizes, int n_in,
                              void* d_out, int out_size, void* d_ws, size_t ws_size,
                              hipStream_t stream)
{
    (void)in_sizes; (void)n_in; (void)out_size; (void)d_ws; (void)ws_size;

    const float* in  = (const float*)d_in[0];
    const int*   idx = (const int*)d_in[1];
    float*       out = (float*)d_out;

    // Exact coverage: (28*4)=112 pixels in W, 64*4B*4=1KB channels per pixel.
    const dim3 block(D4, 4, 1);        // 256 threads = 8 wave32s
    const dim3 grid(W / 4, H, N);      // (28, 112, 8) = 25,088 blocks

    MaxUnPool2DWithIndices_81260781240726_kernel<<<grid, block, 0, stream>>>(in, idx, out);
}